// YOLO_73435350827280
// MI455X (gfx1250) — compile-verified
//
#include <hip/hip_runtime.h>
#include <stdint.h>

// ---- problem constants (B=32, A=5, C=80, H=W=52, stride=32) ----
#define H_    52
#define W_    52
#define HW_   (H_ * W_)        // 2704
#define A_    5
#define C_    80
#define NC5_  (C_ + 5)         // 85
#define CH_   (A_ * NC5_)      // 425
#define FSTRIDE_ 32.0f

#define TILE_HW    32
#define NTHREADS   (TILE_HW * A_)     // 160 threads = 5 wave32s
#define TILE_FLOATS (TILE_HW * CH_)   // 13600 floats = 54.4 KB LDS

#define L2E 1.4426950408889634f

__device__ __forceinline__ float fexp2(float x) {
#if __has_builtin(__builtin_amdgcn_exp2f)
    return __builtin_amdgcn_exp2f(x);      // v_exp_f32
#else
    return exp2f(x);
#endif
}
__device__ __forceinline__ float frcp(float x) {
#if __has_builtin(__builtin_amdgcn_rcpf)
    return __builtin_amdgcn_rcpf(x);       // v_rcp_f32
#else
    return 1.0f / x;
#endif
}
__device__ __forceinline__ float fexp(float x) { return fexp2(x * L2E); }
__device__ __forceinline__ float fsig(float x) { return frcp(1.0f + fexp2(-x * L2E)); }

__global__ void __launch_bounds__(NTHREADS)
yolo_decode_kernel(const float* __restrict__ x,
                   const float* __restrict__ anchors,
                   float* __restrict__ out)
{
    __shared__ float stage[TILE_FLOATS];

    const int tid  = (int)threadIdx.x;
    const int a    = tid >> 5;            // wave id == anchor id (0..4)
    const int lane = tid & 31;            // lane == hw offset inside tile
    const int b    = (int)blockIdx.y;
    const int hw0  = (int)blockIdx.x * TILE_HW;

    const int validHW = (HW_ - hw0 < TILE_HW) ? (HW_ - hw0) : TILE_HW;
    int hw = hw0 + lane;
    if (hw > HW_ - 1) hw = HW_ - 1;       // clamp; clipped at writeback

    // ---- input: lanes step hw -> every one of the 85 channel loads is a
    //      fully-coalesced 128B wave transaction; all loads share one VADDR
    //      (channel byte offsets fit the signed-24b immediate) -------------
    const float* xin = x + (size_t)b * (size_t)(CH_ * HW_)
                         + (size_t)(a * NC5_) * (size_t)HW_ + (size_t)hw;

    float g[5];
#pragma unroll
    for (int c = 0; c < 5; ++c) g[c] = xin[(size_t)c * HW_];

    float cls[C_];
    float m = -3.4e38f;
#pragma unroll
    for (int c = 0; c < C_; ++c) {
        cls[c] = xin[(size_t)(c + 5) * HW_];
        m = fmaxf(m, cls[c]);
    }
    float s = 0.0f;
#pragma unroll
    for (int c = 0; c < C_; ++c) {
        float e = fexp(cls[c] - m);       // stable softmax numerator
        cls[c] = e;
        s += e;
    }
    const float inv = frcp(s);

    // torch meshgrid(arange(W), arange(H), 'ij') quirk: cx = hw/52, cy = hw%52
    const int   cxi = hw / H_;
    const float cxf = (float)cxi;
    const float cyf = (float)(hw - cxi * H_);

    const float pw = anchors[2 * a]     * (1.0f / FSTRIDE_);
    const float ph = anchors[2 * a + 1] * (1.0f / FSTRIDE_);

    // ---- stage result in LDS in OUTPUT layout: (hw_local, a, c) ----------
    // write stride per lane = 425 floats == 41 mod 64 banks (odd -> bijective
    // over lanes) => conflict-free ds_store_b32 pattern.
    float* row = &stage[(lane * A_ + a) * NC5_];
    row[0] = fsig(g[0]) + cxf;
    row[1] = fsig(g[1]) + cyf;
    row[2] = fexp(g[2]) * pw;
    row[3] = fexp(g[3]) * ph;
    row[4] = fsig(g[4]);
#pragma unroll
    for (int c = 0; c < C_; ++c) row[5 + c] = cls[c] * inv;

    __syncthreads();

    // ---- CDNA5 async writeback: LDS -> global directly (no VGPR round
    //      trip), fully coalesced, tracked with ASYNCcnt ------------------
    const float* outTile = out + ((size_t)b * HW_ + (size_t)hw0) * (size_t)CH_;
    const uint64_t outBase = (uint64_t)(uintptr_t)outTile;
    // Flat LDS addresses: low 32 bits == wave-relative LDS byte offset (ISA §10.2)
    const uint32_t ldsBase = (uint32_t)(uintptr_t)(void*)stage;
    const uint32_t validFloats = (uint32_t)validHW * (uint32_t)CH_;

    for (uint32_t i = 0; i < (uint32_t)NC5_; ++i) {      // 85 * 160 == 13600
        uint32_t f = i * (uint32_t)NTHREADS + (uint32_t)tid;
        if (f < validFloats) {
            uint32_t goff  = f * 4u;            // 32-bit VADDR offset (GVS)
            uint32_t laddr = ldsBase + f * 4u;  // LDS byte address (VSRC)
            asm volatile("global_store_async_from_lds_b32 %0, %1, %2"
                         :: "v"(goff), "v"(laddr), "s"(outBase)
                         : "memory");
        }
    }
    asm volatile("s_wait_asynccnt 0" ::: "memory");
}

extern "C" void kernel_launch(void* const* d_in, const int* in_sizes, int n_in,
                              void* d_out, int out_size, void* d_ws, size_t ws_size,
                              hipStream_t stream)
{
    (void)n_in; (void)out_size; (void)d_ws; (void)ws_size;
    const float* x       = (const float*)d_in[0];
    const float* anchors = (const float*)d_in[1];
    float*       out     = (float*)d_out;

    const int B = in_sizes[0] / (CH_ * HW_);          // 32
    dim3 grid((HW_ + TILE_HW - 1) / TILE_HW, B);      // (85, 32)
    yolo_decode_kernel<<<grid, NTHREADS, 0, stream>>>(x, anchors, out);
}